// DiagonalLinear_73581379715571
// MI455X (gfx1250) — compile-verified
//
#include <hip/hip_runtime.h>

// out[b, j] = x[b, j] * clamp(diag[j], -0.95, 0.95)
// BATCH = 16384, LATENT_DIM = 8192, fp32 in / fp32 out.
// Pure streaming kernel: ~1 GiB of traffic, bandwidth-bound (~46 us @ 23.3 TB/s).

typedef float v4f __attribute__((ext_vector_type(4)));

constexpr int kLatent       = 8192;
constexpr int kBatch        = 16384;
constexpr int kTPB          = 256;                    // 8 wave32 waves per block
constexpr int kVec          = 4;                      // float4 per lane -> b128 vmem ops
constexpr int kColsPerBlock = kTPB * kVec;            // 1024 columns per block
constexpr int kGridX        = kLatent / kColsPerBlock; // 8
constexpr int kRowsPerBlock = 64;                     // diagonal reuse per thread
constexpr int kGridY        = kBatch / kRowsPerBlock; // 256

__global__ __launch_bounds__(kTPB) void DiagonalLinear_diag_scale_f32(
    const float* __restrict__ x,
    const float* __restrict__ diag,
    float* __restrict__ out) {
  // Column block: 4 consecutive fp32 columns per lane.
  const int col  = (blockIdx.x * kTPB + threadIdx.x) * kVec;
  const int row0 = blockIdx.y * kRowsPerBlock;

  // Load the 4 diagonal entries once (regular-temporal: tiny, lives in cache),
  // clamp once, reuse for 64 rows.
  v4f d = *reinterpret_cast<const v4f*>(diag + col);
  d.x = fminf(fmaxf(d.x, -0.95f), 0.95f);
  d.y = fminf(fmaxf(d.y, -0.95f), 0.95f);
  d.z = fminf(fmaxf(d.z, -0.95f), 0.95f);
  d.w = fminf(fmaxf(d.w, -0.95f), 0.95f);

  const float* __restrict__ xp = x   + (size_t)row0 * kLatent + col;
  float*       __restrict__ op = out + (size_t)row0 * kLatent + col;

  // Stream 64 rows. Non-temporal on both sides: x/out are touched exactly
  // once (1 GiB total) and must not evict the cached diagonal / thrash L2.
  // Unroll so several b128 loads are outstanding before each wait.
#pragma unroll 8
  for (int r = 0; r < kRowsPerBlock; ++r) {
    v4f v = __builtin_nontemporal_load(reinterpret_cast<const v4f*>(xp));
    v.x *= d.x;
    v.y *= d.y;
    v.z *= d.z;
    v.w *= d.w;
    __builtin_nontemporal_store(v, reinterpret_cast<v4f*>(op));
    xp += kLatent;
    op += kLatent;
  }
}

extern "C" void kernel_launch(void* const* d_in, const int* in_sizes, int n_in,
                              void* d_out, int out_size, void* d_ws, size_t ws_size,
                              hipStream_t stream) {
  const float* x    = (const float*)d_in[0];   // [BATCH, LATENT_DIM] fp32
  const float* diag = (const float*)d_in[1];   // [LATENT_DIM] fp32
  float*       out  = (float*)d_out;           // [BATCH, LATENT_DIM] fp32

  (void)in_sizes; (void)n_in; (void)out_size; (void)d_ws; (void)ws_size;

  dim3 grid(kGridX, kGridY, 1);
  dim3 block(kTPB, 1, 1);
  DiagonalLinear_diag_scale_f32<<<grid, block, 0, stream>>>(x, diag, out);
}